// GCNClassifier_62414464745585
// MI455X (gfx1250) — compile-verified
//
#include <hip/hip_runtime.h>

#define WAVE 32

typedef float v2f __attribute__((ext_vector_type(2)));
typedef float v8f __attribute__((ext_vector_type(8)));

// ---------------- utility device functions ----------------

__device__ __forceinline__ float waveSum(float v) {
#pragma unroll
  for (int o = 16; o > 0; o >>= 1) v += __shfl_xor(v, o, WAVE);
  return v;
}

// order-preserving float<->uint map for atomicMax on floats
__device__ __forceinline__ unsigned f2ord(float f) {
  unsigned u = __float_as_uint(f);
  return (u & 0x80000000u) ? ~u : (u | 0x80000000u);
}
__device__ __forceinline__ float ord2f(unsigned o) {
  unsigned u = (o & 0x80000000u) ? (o & 0x7FFFFFFFu) : ~o;
  return __uint_as_float(u);
}

// ---------------- kernels ----------------

__global__ void k_zerof(float* p, int n) {
  int i = blockIdx.x * blockDim.x + threadIdx.x;
  if (i < n) p[i] = 0.0f;
}

__global__ void k_zerof4(float4* p, int n4) {
  int i = blockIdx.x * blockDim.x + threadIdx.x;
  if (i < n4) p[i] = make_float4(0.f, 0.f, 0.f, 0.f);
}

// L2-normalize each row (wave per row), float4 lane-contiguous loads (F % 4 == 0)
__global__ void k_rownorm(const float* __restrict__ h, float* __restrict__ nh,
                          int N, int F) {
  int wid = blockIdx.x * (blockDim.x >> 5) + (threadIdx.x >> 5);
  int lane = threadIdx.x & 31;
  if (wid >= N) return;
  int F4 = F >> 2;
  const float4* p = (const float4*)(h + (size_t)wid * F);
  float4* q = (float4*)(nh + (size_t)wid * F);
  float acc = 0.0f;
  for (int j = lane; j < F4; j += WAVE) {
    float4 v = p[j];
    acc += v.x * v.x + v.y * v.y + v.z * v.z + v.w * v.w;
  }
  acc = waveSum(acc);
  float inv = 1.0f / fmaxf(sqrtf(acc), 1e-12f);
  for (int j = lane; j < F4; j += WAVE) {
    float4 v = p[j];
    q[j] = make_float4(v.x * inv, v.y * inv, v.z * inv, v.w * inv);
  }
}

// per-edge cosine score (wave per edge) + segment max over dst
__global__ void k_edge_dot(const float* __restrict__ nh, const int* __restrict__ src,
                           const int* __restrict__ dst, float* __restrict__ e,
                           unsigned* __restrict__ emax, const float* __restrict__ beta,
                           float invTemp, int E, int F) {
  int wid = blockIdx.x * (blockDim.x >> 5) + (threadIdx.x >> 5);
  int lane = threadIdx.x & 31;
  if (wid >= E) return;
  int s = src[wid], d = dst[wid];
  int F4 = F >> 2;
  const float4* ps = (const float4*)(nh + (size_t)s * F);
  const float4* pd = (const float4*)(nh + (size_t)d * F);
  float acc = 0.0f;
  for (int j = lane; j < F4; j += WAVE) {
    float4 a = ps[j];
    float4 b = pd[j];
    acc += a.x * b.x + a.y * b.y + a.z * b.z + a.w * b.w;
  }
  acc = waveSum(acc);
  if (lane == 0) {
    float val = beta[0] * acc * invTemp;
    e[wid] = val;
    atomicMax(&emax[d], f2ord(val));
  }
}

// exp(e - max) + segment sum over dst (thread per edge); overwrites e with ex
__global__ void k_edge_exp(float* __restrict__ e, const int* __restrict__ dst,
                           const unsigned* __restrict__ emax, float* __restrict__ esum,
                           int E) {
  int i = blockIdx.x * blockDim.x + threadIdx.x;
  if (i >= E) return;
  int d = dst[i];
  float x = __expf(e[i] - ord2f(emax[d]));
  e[i] = x;
  atomicAdd(&esum[d], x);
}

// weighted scatter-sum of (unnormalized) source features (wave per edge)
__global__ void k_edge_scatter(const float* __restrict__ h, const int* __restrict__ src,
                               const int* __restrict__ dst, const float* __restrict__ ex,
                               const float* __restrict__ esum, float* __restrict__ out,
                               int E, int F) {
  int wid = blockIdx.x * (blockDim.x >> 5) + (threadIdx.x >> 5);
  int lane = threadIdx.x & 31;
  if (wid >= E) return;
  int s = src[wid], d = dst[wid];
  float attn = ex[wid] / esum[d];
  int F4 = F >> 2;
  const float4* ps = (const float4*)(h + (size_t)s * F);
  float* po = out + (size_t)d * F;
  for (int j = lane; j < F4; j += WAVE) {
    float4 v = ps[j];  // one b128 load, four b32 atomics
    atomicAdd(&po[4 * j + 0], v.x * attn);
    atomicAdd(&po[4 * j + 1], v.y * attn);
    atomicAdd(&po[4 * j + 2], v.z * attn);
    atomicAdd(&po[4 * j + 3], v.w * attn);
  }
}

// fea (in place) = conv2 * marker_weight(flat 320); f = sum over 20 markers
__global__ void k_marker(float* __restrict__ fea, const float* __restrict__ mw,
                         float* __restrict__ f, int N) {
  int idx = blockIdx.x * blockDim.x + threadIdx.x;
  if (idx >= N * 16) return;
  int n = idx >> 4, o = idx & 15;
  float s = 0.0f;
#pragma unroll
  for (int m = 0; m < 20; ++m) {
    int k = m * 16 + o;
    float v = fea[(size_t)n * 320 + k] * mw[k];
    fea[(size_t)n * 320 + k] = v;
    s += v;
  }
  f[idx] = s;
}

// BN stats: register accumulate -> LDS ds_add_f32 partials -> 32 global atomics
__global__ void k_bn_stats(const float* __restrict__ f, float* __restrict__ sums, int N) {
  __shared__ float ls[32];
  if (threadIdx.x < 32) ls[threadIdx.x] = 0.0f;
  __syncthreads();
  float acc[16], acq[16];
#pragma unroll
  for (int o = 0; o < 16; ++o) { acc[o] = 0.0f; acq[o] = 0.0f; }
  for (int n = blockIdx.x * blockDim.x + threadIdx.x; n < N; n += gridDim.x * blockDim.x) {
    const float4* row = (const float4*)(f + (size_t)n * 16);
#pragma unroll
    for (int q = 0; q < 4; ++q) {
      float4 v = row[q];
      acc[4 * q + 0] += v.x; acq[4 * q + 0] += v.x * v.x;
      acc[4 * q + 1] += v.y; acq[4 * q + 1] += v.y * v.y;
      acc[4 * q + 2] += v.z; acq[4 * q + 2] += v.z * v.z;
      acc[4 * q + 3] += v.w; acq[4 * q + 3] += v.w * v.w;
    }
  }
#pragma unroll
  for (int o = 0; o < 16; ++o) {
    atomicAdd(&ls[o], acc[o]);
    atomicAdd(&ls[16 + o], acq[o]);
  }
  __syncthreads();
  if (threadIdx.x < 32) atomicAdd(&sums[threadIdx.x], ls[threadIdx.x]);
}

// BN apply (biased var) + LeakyReLU(0.25), in place
__global__ void k_bn_apply(float* __restrict__ f, const float* __restrict__ sums,
                           const float* __restrict__ w, const float* __restrict__ b,
                           int N) {
  int idx = blockIdx.x * blockDim.x + threadIdx.x;
  if (idx >= N * 16) return;
  int o = idx & 15;
  float invN = 1.0f / (float)N;
  float mean = sums[o] * invN;
  float var = sums[16 + o] * invN - mean * mean;
  float v = (f[idx] - mean) * rsqrtf(var + 1e-5f) * w[o] + b[o];
  f[idx] = (v >= 0.0f) ? v : 0.25f * v;
}

// row softmax over 16 channels with temperature (thread per row)
__global__ void k_softmax16(const float* __restrict__ in, float* __restrict__ out,
                            int N, float invT) {
  int n = blockIdx.x * blockDim.x + threadIdx.x;
  if (n >= N) return;
  float v[16];
  const float4* row = (const float4*)(in + (size_t)n * 16);
#pragma unroll
  for (int q = 0; q < 4; ++q) {
    float4 t = row[q];
    v[4 * q + 0] = t.x * invT;
    v[4 * q + 1] = t.y * invT;
    v[4 * q + 2] = t.z * invT;
    v[4 * q + 3] = t.w * invT;
  }
  float mx = -3.4e38f;
#pragma unroll
  for (int o = 0; o < 16; ++o) mx = fmaxf(mx, v[o]);
  float s = 0.0f;
#pragma unroll
  for (int o = 0; o < 16; ++o) {
    v[o] = __expf(v[o] - mx);
    s += v[o];
  }
  float inv = 1.0f / s;
  float4* orow = (float4*)(out + (size_t)n * 16);
#pragma unroll
  for (int q = 0; q < 4; ++q)
    orow[q] = make_float4(v[4 * q] * inv, v[4 * q + 1] * inv, v[4 * q + 2] * inv,
                          v[4 * q + 3] * inv);
}

// P[N,320] = S[N,16] @ Bm[16,320] via V_WMMA_F32_16X16X4_F32 (wave per 16x16 tile)
__global__ void k_gemm_wmma(const float* __restrict__ S, const float* __restrict__ Bm,
                            float* __restrict__ P, int N) {
  const int tilesN = 320 / 16;  // 20
  int wave = blockIdx.x * (blockDim.x >> 5) + (threadIdx.x >> 5);
  int lane = threadIdx.x & 31;
  int tileM = wave / tilesN;
  int tileN = wave % tilesN;
  int row0 = tileM * 16;
  if (row0 >= N) return;

  int m = lane & 15;
  int half = lane >> 4;                 // 0: lanes 0-15, 1: lanes 16-31
  int row = min(row0 + m, N - 1);       // clamp: garbage rows never stored
  int col = tileN * 16 + m;             // B-fragment column
  int kh = half * 2;                    // K sub-offset within 4-wide step

  const v2f* arow = (const v2f*)(S + (size_t)row * 16 + kh);  // b64 loads
  v8f c = {};
#pragma unroll
  for (int k0 = 0; k0 < 16; k0 += 4) {
    v2f a = arow[k0 >> 1];  // floats [row*16 + k0 + kh], [.. + 1]
    v2f b;
    b.x = Bm[(size_t)(k0 + kh) * 320 + col];
    b.y = Bm[(size_t)(k0 + kh + 1) * 320 + col];
    c = __builtin_amdgcn_wmma_f32_16x16x4_f32(false, a, false, b, (short)0, c,
                                              false, false);
  }
  // C/D layout: VGPR i -> M = i + 8*half, N = lane&15
  int baseRow = row0 + half * 8;
  float* pout = P + (size_t)baseRow * 320 + tileN * 16 + m;
  if (row0 + 16 <= N) {  // full tile: unconditional stores
#pragma unroll
    for (int i = 0; i < 8; ++i) pout[(size_t)i * 320] = c[i];
  } else {
#pragma unroll
    for (int i = 0; i < 8; ++i)
      if (baseRow + i < N) pout[(size_t)i * 320] = c[i];
  }
}

// ---------------- host side ----------------

static inline int ceil_div(int a, int b) { return (a + b - 1) / b; }

static void launch_agnn(const float* h, float* nh_buf, const int* src, const int* dst,
                        float* e_buf, unsigned* emax, float* esum, const float* beta,
                        float* out, int N, int E, int F, hipStream_t stream) {
  const int B = 256;
  const int WPB = B / 32;
  k_rownorm<<<ceil_div(N, WPB), B, 0, stream>>>(h, nh_buf, N, F);
  k_zerof<<<ceil_div(N, B), B, 0, stream>>>((float*)emax, N);
  k_zerof<<<ceil_div(N, B), B, 0, stream>>>(esum, N);
  k_edge_dot<<<ceil_div(E, WPB), B, 0, stream>>>(nh_buf, src, dst, e_buf, emax, beta,
                                                 1.0f, E, F);
  k_edge_exp<<<ceil_div(E, B), B, 0, stream>>>(e_buf, dst, emax, esum, E);
  k_zerof4<<<ceil_div(N * (F >> 2), B), B, 0, stream>>>((float4*)out, N * (F >> 2));
  k_edge_scatter<<<ceil_div(E, WPB), B, 0, stream>>>(h, src, dst, e_buf, esum, out, E, F);
}

extern "C" void kernel_launch(void* const* d_in, const int* in_sizes, int n_in,
                              void* d_out, int out_size, void* d_ws, size_t ws_size,
                              hipStream_t stream) {
  const float* x = (const float*)d_in[0];       // [N,320]
  const int* src = (const int*)d_in[1];         // [E]
  const int* dst = (const int*)d_in[2];         // [E]
  const float* mw = (const float*)d_in[3];      // [20,16] flat 320
  const float* bn_w = (const float*)d_in[4];    // [16]
  const float* bn_b = (const float*)d_in[5];    // [16]
  const float* refc = (const float*)d_in[6];    // [16,320]
  const float* beta = (const float*)d_in[7];    // [1]

  const int FEAT = 320, OUT = 16;
  const int N = in_sizes[0] / FEAT;
  const int E = in_sizes[1];

  // d_out layout: fea [N,320] | f [N,16] | p [N,320] | out [N,16]
  float* out_fea = (float*)d_out;
  float* out_f = out_fea + (size_t)N * FEAT;
  float* out_p = out_f + (size_t)N * OUT;
  float* out_o = out_p + (size_t)N * FEAT;

  // workspace layout
  float* wsA = (float*)d_ws;                       // N*320 (nh / conv1 result)
  float* e_buf = wsA + (size_t)N * FEAT;           // E
  unsigned* emax = (unsigned*)(e_buf + E);         // N
  float* esum = (float*)(emax + N);                // N
  float* nf = esum + N;                            // N*16
  float* g1 = nf + (size_t)N * OUT;                // N*16
  float* g2 = g1 + (size_t)N * OUT;                // N*16
  float* smax = g2 + (size_t)N * OUT;              // N*16
  float* bnsums = smax + (size_t)N * OUT;          // 32

  const int B = 256;
  const int WPB = B / 32;

  // ---- conv1 on x (F=320): nh in wsA, then scatter result overwrites wsA ----
  k_rownorm<<<ceil_div(N, WPB), B, 0, stream>>>(x, wsA, N, FEAT);
  k_zerof<<<ceil_div(N, B), B, 0, stream>>>((float*)emax, N);
  k_zerof<<<ceil_div(N, B), B, 0, stream>>>(esum, N);
  k_edge_dot<<<ceil_div(E, WPB), B, 0, stream>>>(wsA, src, dst, e_buf, emax, beta, 1.0f,
                                                 E, FEAT);
  k_edge_exp<<<ceil_div(E, B), B, 0, stream>>>(e_buf, dst, emax, esum, E);
  k_zerof4<<<ceil_div(N * (FEAT >> 2), B), B, 0, stream>>>((float4*)wsA,
                                                           N * (FEAT >> 2));
  k_edge_scatter<<<ceil_div(E, WPB), B, 0, stream>>>(x, src, dst, e_buf, esum, wsA, E,
                                                     FEAT);

  // ---- conv2 (F=320): nh2 borrows d_out p-region (fully overwritten later) ----
  launch_agnn(wsA, out_p, src, dst, e_buf, emax, esum, beta, out_fea, N, E, FEAT,
              stream);

  // ---- marker weighting (in place on fea) + f = channel sums ----
  k_marker<<<ceil_div(N * OUT, B), B, 0, stream>>>(out_fea, mw, out_f, N);

  // ---- BatchNorm (batch stats) + LeakyReLU(0.25) on f ----
  k_zerof<<<1, 32, 0, stream>>>(bnsums, 32);
  int statBlocks = min(512, ceil_div(N, B));
  k_bn_stats<<<statBlocks, B, 0, stream>>>(out_f, bnsums, N);
  k_bn_apply<<<ceil_div(N * OUT, B), B, 0, stream>>>(out_f, bnsums, bn_w, bn_b, N);

  // ---- p = softmax(f/0.1) @ ref_center via WMMA f32 16x16x4 ----
  k_softmax16<<<ceil_div(N, B), B, 0, stream>>>(out_f, smax, N, 10.0f);
  int tiles = ceil_div(N, 16) * (FEAT / 16);
  k_gemm_wmma<<<ceil_div(tiles, WPB), B, 0, stream>>>(smax, refc, out_p, N);

  // ---- conv3 + conv4 on f (F=16) ----
  launch_agnn(out_f, nf, src, dst, e_buf, emax, esum, beta, g1, N, E, OUT, stream);
  launch_agnn(g1, nf, src, dst, e_buf, emax, esum, beta, g2, N, E, OUT, stream);

  // ---- out = softmax(conv4 / 5) ----
  k_softmax16<<<ceil_div(N, B), B, 0, stream>>>(g2, out_o, N, 0.2f);
}